// SpectralConv1d_56710748177057
// MI455X (gfx1250) — compile-verified
//
#include <hip/hip_runtime.h>

typedef float v2f __attribute__((ext_vector_type(2)));
typedef float v8f __attribute__((ext_vector_type(8)));

#define NMODES 32
#define NN     8192      // signal length
#define CIN    64
#define COUT   64
#define BATCH  32
#define ROWS   (BATCH * CIN)   // 2048
#define KCOEF  64              // 32 cos + 32 sin coefficients
#define KSPLIT 16              // K-chunks for stage-1 parallelism
#define KCHUNK (NN / KSPLIT)   // 512

// workspace layout (in floats)
#define OFF_BAS  0                           // [64][8192]  basis, j-major (cos rows 0..31, sin rows 32..63)
#define OFF_BAST (OFF_BAS  + KCOEF * NN)     // [8192][64]  basis transposed, n-major
#define OFF_C1P  (OFF_BAST + NN * KCOEF)     // [KSPLIT][2048][64] stage-1 partials
#define OFF_C1   (OFF_C1P  + KSPLIT * ROWS * KCOEF)  // [2048][64]
#define OFF_A3   (OFF_C1   + ROWS * KCOEF)           // [2048][64]

// ---------------------------------------------------------------------------
// Basis: BAS[j][n] = cos(2*pi*m*n/N) for j=m<32, sin(2*pi*m*n/N) for j=32+m.
// Exact integer phase reduction keeps sincos args in [0, 2*pi).
// ---------------------------------------------------------------------------
__global__ void fno_basis_kernel(float* __restrict__ ws) {
    int idx = blockIdx.x * blockDim.x + threadIdx.x;   // 0 .. 64*8192-1
    int j = idx & 63;
    int n = idx >> 6;
    int m = j & 31;
    int r = (m * n) & (NN - 1);
    float theta = (float)r * (6.28318530717958647692f / (float)NN);
    float s, c;
    __sincosf(theta, &s, &c);
    float v = (j < 32) ? c : s;
    ws[OFF_BAS  + (size_t)j * NN + n] = v;
    ws[OFF_BAST + (size_t)n * KCOEF + j] = v;
}

// ---------------------------------------------------------------------------
// Stage 1: truncated forward DFT as fp32 WMMA GEMM.
// C1[row][j] = sum_n x[row][n] * BAS[j][n],  row = b*64+i.
// One wave = 16-row strip x all 64 cols (4 accumulators); K split KSPLIT ways.
// ---------------------------------------------------------------------------
__global__ void fno_dft_kernel(const float* __restrict__ x, float* __restrict__ ws) {
    const float* bas = ws + OFF_BAS;
    float* c1p = ws + OFF_C1P;

    int lane  = threadIdx.x & 31;
    int wave  = threadIdx.x >> 5;
    int strip = blockIdx.x * 8 + wave;       // 0..127
    int chunk = blockIdx.y;                  // 0..KSPLIT-1
    int row0  = strip * 16;
    int mrow  = lane & 15;                   // A: row in tile / B: col in tile
    int koff  = (lane >> 4) << 1;            // K-pair select

    const float* xrow = x + (size_t)(row0 + mrow) * NN;
    const float* b0p = bas + (size_t)(0 * 16 + mrow) * NN;
    const float* b1p = bas + (size_t)(1 * 16 + mrow) * NN;
    const float* b2p = bas + (size_t)(2 * 16 + mrow) * NN;
    const float* b3p = bas + (size_t)(3 * 16 + mrow) * NN;

    v8f acc0 = {}, acc1 = {}, acc2 = {}, acc3 = {};
    int kbeg = chunk * KCHUNK;
    #pragma unroll 4
    for (int k0 = kbeg; k0 < kbeg + KCHUNK; k0 += 4) {
        int ka = k0 + koff;
        v2f a  = *(const v2f*)(xrow + ka);
        v2f b0 = *(const v2f*)(b0p + ka);
        v2f b1 = *(const v2f*)(b1p + ka);
        v2f b2 = *(const v2f*)(b2p + ka);
        v2f b3 = *(const v2f*)(b3p + ka);
        acc0 = __builtin_amdgcn_wmma_f32_16x16x4_f32(false, a, false, b0, (short)0, acc0, false, false);
        acc1 = __builtin_amdgcn_wmma_f32_16x16x4_f32(false, a, false, b1, (short)0, acc1, false, false);
        acc2 = __builtin_amdgcn_wmma_f32_16x16x4_f32(false, a, false, b2, (short)0, acc2, false, false);
        acc3 = __builtin_amdgcn_wmma_f32_16x16x4_f32(false, a, false, b3, (short)0, acc3, false, false);
    }

    float* outp = c1p + (size_t)chunk * ROWS * KCOEF;
    int drow = row0 + ((lane >> 4) << 3);
    #pragma unroll
    for (int r = 0; r < 8; ++r) {
        outp[(size_t)(drow + r) * KCOEF +  0 + mrow] = acc0[r];
        outp[(size_t)(drow + r) * KCOEF + 16 + mrow] = acc1[r];
        outp[(size_t)(drow + r) * KCOEF + 32 + mrow] = acc2[r];
        outp[(size_t)(drow + r) * KCOEF + 48 + mrow] = acc3[r];
    }
}

// Deterministic reduction of the K-split partials.
__global__ void fno_reduce_kernel(float* __restrict__ ws) {
    const float* c1p = ws + OFF_C1P;
    float* c1 = ws + OFF_C1;
    int idx = blockIdx.x * blockDim.x + threadIdx.x;   // 0 .. ROWS*KCOEF-1
    float acc = 0.0f;
    #pragma unroll
    for (int s = 0; s < KSPLIT; ++s)
        acc += c1p[(size_t)s * ROWS * KCOEF + idx];
    c1[idx] = acc;
}

// ---------------------------------------------------------------------------
// Stage 2: complex mode mix over Cin + irfft scaling folded into A3.
//   Xr = C1[row][m], Xi = -C1[row][32+m]  (stage 1 summed +sin)
//   Or = Xr*Wr - Xi*Wi ;  Oi = Xr*Wi + Xi*Wr
//   A3[:,m]    = (m==0 ? Or0 : 2*Or_m) / N
//   A3[:,32+m] = (m==0 ? 0   : -2*Oi_m) / N   (irfft drops Im of DC bin)
// ---------------------------------------------------------------------------
__global__ void fno_mix_kernel(const float* __restrict__ wr, const float* __restrict__ wi,
                               float* __restrict__ ws) {
    const float* c1 = ws + OFF_C1;
    float* a3 = ws + OFF_A3;
    int idx = blockIdx.x * blockDim.x + threadIdx.x;   // 0 .. ROWS*64-1
    int j   = idx & 63;
    int row = idx >> 6;      // b*64 + o
    int b   = row >> 6;
    int o   = row & 63;
    int m   = j & 31;
    bool imagpart = (j >= 32);

    float acc = 0.0f;
    for (int i = 0; i < CIN; ++i) {
        const float* crow = c1 + (size_t)(b * CIN + i) * KCOEF;
        float xr  =  crow[m];
        float xi  = -crow[32 + m];
        float wre = wr[(size_t)(i * COUT + o) * NMODES + m];
        float wim = wi[(size_t)(i * COUT + o) * NMODES + m];
        acc += imagpart ? (xr * wim + xi * wre) : (xr * wre - xi * wim);
    }
    const float invn = 1.0f / (float)NN;
    float v;
    if (j == 0)          v = acc * invn;
    else if (j == 32)    v = 0.0f;
    else if (!imagpart)  v = acc * (2.0f * invn);
    else                 v = -acc * (2.0f * invn);
    a3[(size_t)row * KCOEF + j] = v;
}

// ---------------------------------------------------------------------------
// Stage 3: sparse irfft as fp32 WMMA GEMM.
// out[row][n] = sum_j A3[row][j] * BAS[j][n].  A-fragments (16x64) cached in
// VGPRs and reused across 8 column tiles per wave.
// ---------------------------------------------------------------------------
__global__ void fno_irfft_kernel(const float* __restrict__ ws, float* __restrict__ out) {
    const float* bast = ws + OFF_BAST;
    const float* a3   = ws + OFF_A3;

    int lane  = threadIdx.x & 31;
    int wave  = threadIdx.x >> 5;
    int row0  = blockIdx.x * 16;             // 128 row strips
    int mrow  = lane & 15;
    int koff  = (lane >> 4) << 1;

    v2f af[16];
    const float* arow = a3 + (size_t)(row0 + mrow) * KCOEF;
    #pragma unroll
    for (int kk = 0; kk < 16; ++kk)
        af[kk] = *(const v2f*)(arow + kk * 4 + koff);

    int colbase = (blockIdx.y * 8 + wave) * 128;   // 8 waves * 8 col-tiles * 16
    int drow = row0 + ((lane >> 4) << 3);
    for (int ct = 0; ct < 8; ++ct) {
        int col = colbase + ct * 16 + mrow;
        const float* bcol = bast + (size_t)col * KCOEF;
        v8f acc = {};
        #pragma unroll
        for (int kk = 0; kk < 16; ++kk) {
            v2f bv = *(const v2f*)(bcol + kk * 4 + koff);
            acc = __builtin_amdgcn_wmma_f32_16x16x4_f32(false, af[kk], false, bv, (short)0, acc, false, false);
        }
        #pragma unroll
        for (int r = 0; r < 8; ++r)
            out[(size_t)(drow + r) * NN + col] = acc[r];
    }
}

// ---------------------------------------------------------------------------
extern "C" void kernel_launch(void* const* d_in, const int* in_sizes, int n_in,
                              void* d_out, int out_size, void* d_ws, size_t ws_size,
                              hipStream_t stream) {
    const float* x  = (const float*)d_in[0];   // (32, 64, 8192)
    const float* wr = (const float*)d_in[1];   // (64, 64, 32)
    const float* wi = (const float*)d_in[2];   // (64, 64, 32)
    float* out = (float*)d_out;                // (32, 64, 8192)
    float* ws  = (float*)d_ws;

    // basis: 64*8192 elements
    fno_basis_kernel<<<(KCOEF * NN) / 256, 256, 0, stream>>>(ws);
    // stage 1: 128 strips (16 blocks x 8 waves) x KSPLIT K-chunks
    fno_dft_kernel<<<dim3(16, KSPLIT), 256, 0, stream>>>(x, ws);
    // reduce partials: ROWS*KCOEF elements
    fno_reduce_kernel<<<(ROWS * KCOEF) / 256, 256, 0, stream>>>(ws);
    // mode mix: ROWS*64 threads
    fno_mix_kernel<<<(ROWS * KCOEF) / 256, 256, 0, stream>>>(wr, wi, ws);
    // stage 3: 128 row strips x (8 blocks x 8 waves) column groups
    fno_irfft_kernel<<<dim3(128, 8), 256, 0, stream>>>(ws, out);
}